// Net_27273042330406
// MI455X (gfx1250) — compile-verified
//
#include <hip/hip_runtime.h>
#include <hip/hip_bf16.h>
#include <math.h>

typedef float v2f __attribute__((ext_vector_type(2)));
typedef float v8f __attribute__((ext_vector_type(8)));

#define LRELU_FEAT 0.01f
#define LRELU_ATT  0.2f

// ---------------------------------------------------------------------------
// Order-preserving float<->uint mapping for atomicMax-based segment max.
// ---------------------------------------------------------------------------
__device__ __forceinline__ unsigned ordkey(float f) {
  unsigned u = __float_as_uint(f);
  return (u & 0x80000000u) ? ~u : (u | 0x80000000u);
}
__device__ __forceinline__ float ordunkey_guard(unsigned k) {
  unsigned u = (k & 0x80000000u) ? (k & 0x7fffffffu) : ~k;
  if ((u & 0x7f800000u) == 0x7f800000u) return 0.f;  // -inf sentinel / nan -> 0
  return __uint_as_float(u);
}

__device__ __forceinline__ void edge_sd(int j, int E, int doubled,
                                        const int* __restrict__ src,
                                        const int* __restrict__ dst,
                                        int& s, int& d) {
  if (!doubled || j < E) { s = src[j]; d = dst[j]; }
  else                   { s = dst[j - E]; d = src[j - E]; }
}

// ---------------------------------------------------------------------------
// WMMA fp32 GEMM:  C[M x 64] = act( (accum ? C : 0) + A[M x 64|128] @ W[K x 64] + bias )
// One wave computes a 16x16 tile via V_WMMA_F32_16X16X4_F32 (K-step 4).
// Block = 128 threads = 4 waves = full 64-column strip of one 16-row tile.
// M must be a multiple of 16 (true here: 30000, 60000).
// ---------------------------------------------------------------------------
template<int K>
__global__ void __launch_bounds__(128)
gemm_wmma_kernel(const float* __restrict__ A, const float* __restrict__ W,
                 const float* __restrict__ bias, float* __restrict__ C,
                 int accum, int act)
{
  const int row0 = blockIdx.x * 16;
  const int wave = threadIdx.x >> 5;
  const int lane = threadIdx.x & 31;
  const int half = lane >> 4;     // 0: lanes 0-15, 1: lanes 16-31
  const int lm   = lane & 15;
  const int col0 = wave * 16;

  v8f c = {0.f, 0.f, 0.f, 0.f, 0.f, 0.f, 0.f, 0.f};
  if (accum) {
#pragma unroll
    for (int j = 0; j < 8; ++j)
      c[j] = C[(row0 + j + 8 * half) * 64 + col0 + lm];
  }

  const float* arow = A + (row0 + lm) * K;
#pragma unroll
  for (int k0 = 0; k0 < K; k0 += 4) {
    // A 16x4 fp32: VGPR j holds K = k0 + 2*half + j for row M = lm
    v2f a, b;
    a[0] = arow[k0 + 2 * half + 0];
    a[1] = arow[k0 + 2 * half + 1];
    // B 4x16 fp32: VGPR j holds K-row (k0 + 2*half + j), col N = lm
    b[0] = W[(k0 + 2 * half + 0) * 64 + col0 + lm];
    b[1] = W[(k0 + 2 * half + 1) * 64 + col0 + lm];
    c = __builtin_amdgcn_wmma_f32_16x16x4_f32(false, a, false, b, (short)0, c,
                                              false, false);
  }

  const float bv = bias ? bias[col0 + lm] : 0.f;
#pragma unroll
  for (int j = 0; j < 8; ++j) {
    float v = c[j] + bv;
    if (act) v = v > 0.f ? v : LRELU_FEAT * v;
    C[(row0 + j + 8 * half) * 64 + col0 + lm] = v;
  }
}

// ---------------------------------------------------------------------------
// Elementwise / row-wise helpers (wave32-aware: one wave per 64-wide row).
// ---------------------------------------------------------------------------
__global__ void fill_zero_kernel(float* __restrict__ p, long long n) {
  long long i = (long long)blockIdx.x * blockDim.x + threadIdx.x;
  if (i < n) p[i] = 0.f;
}

__global__ void l2norm_kernel(const float* __restrict__ src, float* __restrict__ dst,
                              int rows) {
  int wave = (int)(((long long)blockIdx.x * blockDim.x + threadIdx.x) >> 5);
  int lane = threadIdx.x & 31;
  if (wave >= rows) return;
  const float* r = src + (size_t)wave * 64;
  float x0 = r[lane], x1 = r[lane + 32];
  float s = x0 * x0 + x1 * x1;
#pragma unroll
  for (int off = 16; off; off >>= 1) s += __shfl_xor(s, off, 32);
  float inv = 1.f / fmaxf(sqrtf(s), 1e-12f);
  float* w = dst + (size_t)wave * 64;
  w[lane] = x0 * inv;
  w[lane + 32] = x1 * inv;
}

// pref <- l2norm(pref + gat_out + bias), rows 0..rows-1 of x (in place)
__global__ void pref_update_kernel(float* __restrict__ x, const float* __restrict__ o,
                                   const float* __restrict__ bg, int rows) {
  int wave = (int)(((long long)blockIdx.x * blockDim.x + threadIdx.x) >> 5);
  int lane = threadIdx.x & 31;
  if (wave >= rows) return;
  size_t base = (size_t)wave * 64;
  float v0 = x[base + lane]      + o[base + lane]      + bg[lane];
  float v1 = x[base + lane + 32] + o[base + lane + 32] + bg[lane + 32];
  float s = v0 * v0 + v1 * v1;
#pragma unroll
  for (int off = 16; off; off >>= 1) s += __shfl_xor(s, off, 32);
  float inv = 1.f / fmaxf(sqrtf(s), 1e-12f);
  x[base + lane] = v0 * inv;
  x[base + lane + 32] = v1 * inv;
}

// hs[n] = h[n].asrc ; hd[n] = h[n].adst   (one wave per row)
__global__ void rowdot2_kernel(const float* __restrict__ h, const float* __restrict__ asrc,
                               const float* __restrict__ adst, float* __restrict__ hs,
                               float* __restrict__ hd, int rows) {
  int wave = (int)(((long long)blockIdx.x * blockDim.x + threadIdx.x) >> 5);
  int lane = threadIdx.x & 31;
  if (wave >= rows) return;
  const float* r = h + (size_t)wave * 64;
  float x0 = r[lane], x1 = r[lane + 32];
  float s = x0 * asrc[lane] + x1 * asrc[lane + 32];
  float d = x0 * adst[lane] + x1 * adst[lane + 32];
#pragma unroll
  for (int off = 16; off; off >>= 1) {
    s += __shfl_xor(s, off, 32);
    d += __shfl_xor(d, off, 32);
  }
  if (lane == 0) { hs[wave] = s; hd[wave] = d; }
}

// ---------------------------------------------------------------------------
// GAT edge kernels
// ---------------------------------------------------------------------------
__global__ void edge_att_kernel(const float* __restrict__ hs, const float* __restrict__ hd,
                                const int* __restrict__ src, const int* __restrict__ dst,
                                int E, int doubled, float* __restrict__ ebuf,
                                unsigned* mkeys) {
  int total = doubled ? 2 * E : E;
  int j = blockIdx.x * blockDim.x + threadIdx.x;
  if (j >= total) return;
  int s, d; edge_sd(j, E, doubled, src, dst, s, d);
  float e = hs[s] + hd[d];
  e = e > 0.f ? e : LRELU_ATT * e;
  ebuf[j] = e;
  atomicMax(&mkeys[d], ordkey(e));
}

// decode keys -> float max (0 if segment empty), zero denom
__global__ void node_softmax_prep_kernel(unsigned* keys, float* mout,
                                         float* __restrict__ denom, int n) {
  int i = blockIdx.x * blockDim.x + threadIdx.x;
  if (i >= n) return;
  float m = ordunkey_guard(keys[i]);
  mout[i] = m;       // same buffer as keys (reinterpreted), read-then-write
  denom[i] = 0.f;
}

__global__ void edge_exp_kernel(float* __restrict__ ebuf, const float* __restrict__ mbuf,
                                float* denom, const int* __restrict__ src,
                                const int* __restrict__ dst, int E, int doubled) {
  int total = doubled ? 2 * E : E;
  int j = blockIdx.x * blockDim.x + threadIdx.x;
  if (j >= total) return;
  int s, d; edge_sd(j, E, doubled, src, dst, s, d);
  float ex = expf(ebuf[j] - mbuf[d]);
  ebuf[j] = ex;
  atomicAdd(&denom[d], ex);
}

__global__ void edge_alpha_kernel(const float* __restrict__ ebuf,
                                  const float* __restrict__ denom,
                                  const int* __restrict__ src, const int* __restrict__ dst,
                                  int E, int doubled, float* __restrict__ alpha) {
  int total = doubled ? 2 * E : E;
  int j = blockIdx.x * blockDim.x + threadIdx.x;
  if (j >= total) return;
  int s, d; edge_sd(j, E, doubled, src, dst, s, d);
  alpha[j] = ebuf[j] / (denom[d] + 1e-16f);
}

// out[dst] += h[src] * alpha   (64 threads per edge)
__global__ void scatter_att_kernel(const float* __restrict__ h,
                                   const float* __restrict__ alpha,
                                   const int* __restrict__ src, const int* __restrict__ dst,
                                   int E, int doubled, float* out) {
  long long total = (long long)(doubled ? 2 * E : E) * 64;
  long long i = (long long)blockIdx.x * blockDim.x + threadIdx.x;
  if (i >= total) return;
  int j = (int)(i >> 6), c = (int)(i & 63);
  int s, d; edge_sd(j, E, doubled, src, dst, s, d);
  atomicAdd(&out[(size_t)d * 64 + c], h[(size_t)s * 64 + c] * alpha[j]);
}

// rep = x + leaky_relu(gat_out + bias, 0.01)
__global__ void rep_kernel(const float* __restrict__ x, const float* __restrict__ o,
                           const float* __restrict__ bg, float* __restrict__ rep,
                           long long n64) {
  long long i = (long long)blockIdx.x * blockDim.x + threadIdx.x;
  if (i >= n64) return;
  int c = (int)(i & 63);
  float t = o[i] + bg[c];
  t = t > 0.f ? t : LRELU_FEAT * t;
  rep[i] = x[i] + t;
}

// ---------------------------------------------------------------------------
// Fusion weight + SAGE scatter
// ---------------------------------------------------------------------------
__global__ void weight_kernel(const float* __restrict__ av, const float* __restrict__ aa,
                              const float* __restrict__ msc, const int* __restrict__ src,
                              const int* __restrict__ dst, int E, float* __restrict__ w) {
  int j = blockIdx.x * blockDim.x + threadIdx.x;
  if (j >= 2 * E) return;
  int node = (j < E) ? src[j] : dst[j - E];
  float m = fmaxf(av[j] * msc[(size_t)node * 2 + 0], aa[j] * msc[(size_t)node * 2 + 1]);
  w[j] = fmaxf(m, 0.f);
}

// agg[dst2] += x[src2] * w  over doubled edges (64 threads per edge)
__global__ void scatter_sage_kernel(const float* __restrict__ x,
                                    const float* __restrict__ w,
                                    const int* __restrict__ src, const int* __restrict__ dst,
                                    int E, float* agg) {
  long long total = (long long)2 * E * 64;
  long long i = (long long)blockIdx.x * blockDim.x + threadIdx.x;
  if (i >= total) return;
  int j = (int)(i >> 6), c = (int)(i & 63);
  int s, d; edge_sd(j, E, 1, src, dst, s, d);
  atomicAdd(&agg[(size_t)d * 64 + c], x[(size_t)s * 64 + c] * w[j]);
}

// out[n, 0:64]=xn+h1+h2 ; out[n,64:128]=rep_v ; out[n,128:192]=rep_a
__global__ void final_out_kernel(const float* __restrict__ xn, const float* __restrict__ h1,
                                 const float* __restrict__ h2, const float* __restrict__ rv,
                                 const float* __restrict__ ra, float* __restrict__ out,
                                 long long n64) {
  long long i = (long long)blockIdx.x * blockDim.x + threadIdx.x;
  if (i >= n64) return;
  long long n = i >> 6;
  int c = (int)(i & 63);
  out[n * 192 + c]       = xn[i] + h1[i] + h2[i];
  out[n * 192 + 64 + c]  = rv[i];
  out[n * 192 + 128 + c] = ra[i];
}

// ---------------------------------------------------------------------------
// Host orchestration
// ---------------------------------------------------------------------------
extern "C" void kernel_launch(void* const* d_in, const int* in_sizes, int n_in,
                              void* d_out, int out_size, void* d_ws, size_t ws_size,
                              hipStream_t stream) {
  const float* v_feat = (const float*)d_in[0];
  const float* a_feat = (const float*)d_in[1];
  const float* pref_v = (const float*)d_in[2];
  const float* pref_a = (const float*)d_in[3];
  const float* Wf_v   = (const float*)d_in[4];
  const float* bf_v   = (const float*)d_in[5];
  const float* Wf_a   = (const float*)d_in[6];
  const float* bf_a   = (const float*)d_in[7];
  const float* Wg_v   = (const float*)d_in[8];
  const float* bg_v   = (const float*)d_in[9];
  const float* asrc_v = (const float*)d_in[10];
  const float* adst_v = (const float*)d_in[11];
  const float* Wg_a   = (const float*)d_in[12];
  const float* bg_a   = (const float*)d_in[13];
  const float* asrc_a = (const float*)d_in[14];
  const float* adst_a = (const float*)d_in[15];
  const float* id_emb = (const float*)d_in[16];
  const float* W1l    = (const float*)d_in[17];
  const float* W1r    = (const float*)d_in[18];
  const float* b1     = (const float*)d_in[19];
  const float* W2l    = (const float*)d_in[20];
  const float* W2r    = (const float*)d_in[21];
  const float* b2     = (const float*)d_in[22];
  const float* msc    = (const float*)d_in[23];
  const int*   src    = (const int*)d_in[24];
  const int*   dst    = (const int*)d_in[25];
  (void)n_in; (void)out_size; (void)ws_size;

  const int NU = in_sizes[2] / 64;    // 30000
  const int N  = in_sizes[16] / 64;   // 60000
  const int NI = N - NU;              // 30000
  const int E  = in_sizes[24];        // 600000

  // ---- workspace layout ----
  char* wsp = (char*)d_ws;
  size_t off = 0;
  auto alloc = [&](size_t bytes) -> float* {
    float* p = (float*)(wsp + off);
    off += (bytes + 255) & ~(size_t)255;
    return p;
  };
  float* xbuf   = alloc((size_t)N * 64 * 4);   // node features x (pref ; feats) / xn
  float* hbuf   = alloc((size_t)N * 64 * 4);   // h = x @ Wg / h2 scratch
  float* obuf   = alloc((size_t)N * 64 * 4);   // gat out / sage agg
  float* h1buf  = alloc((size_t)N * 64 * 4);   // egcn h1
  float* repv   = alloc((size_t)N * 64 * 4);
  float* repa   = alloc((size_t)N * 64 * 4);
  float* hsb    = alloc((size_t)N * 4);
  float* hdb    = alloc((size_t)N * 4);
  float* mbuf   = alloc((size_t)N * 4);        // max keys then float max
  float* denom  = alloc((size_t)N * 4);
  float* ebuf   = alloc((size_t)2 * E * 4);    // edge e / ex / routing alpha
  float* alphav = alloc((size_t)2 * E * 4);
  float* alphaa = alloc((size_t)2 * E * 4);
  float* wedge  = alloc((size_t)2 * E * 4);

  const int TPB = 256;
  auto blocks = [&](long long n) { return (unsigned)((n + TPB - 1) / TPB); };

  auto gemm64 = [&](const float* A, const float* W, const float* bias, float* C,
                    int M, int accum, int act) {
    gemm_wmma_kernel<64><<<M / 16, 128, 0, stream>>>(A, W, bias, C, accum, act);
  };
  auto gemm128 = [&](const float* A, const float* W, const float* bias, float* C,
                     int M, int accum, int act) {
    gemm_wmma_kernel<128><<<M / 16, 128, 0, stream>>>(A, W, bias, C, accum, act);
  };

  // One full GAT message pass over current xbuf. Result in obuf; alpha in alpha_dst.
  auto gat_pass = [&](const float* Wg, const float* bg, const float* asrc,
                      const float* adst, int doubled, float* alpha_dst) {
    const int TE = doubled ? 2 * E : E;
    gemm64(xbuf, Wg, nullptr, hbuf, N, 0, 0);                      // h = x @ Wg
    rowdot2_kernel<<<(N + 7) / 8, 256, 0, stream>>>(hbuf, asrc, adst, hsb, hdb, N);
    fill_zero_kernel<<<blocks(N), TPB, 0, stream>>>(mbuf, N);      // key 0 == -inf sentinel
    edge_att_kernel<<<blocks(TE), TPB, 0, stream>>>(hsb, hdb, src, dst, E, doubled,
                                                    ebuf, (unsigned*)mbuf);
    node_softmax_prep_kernel<<<blocks(N), TPB, 0, stream>>>((unsigned*)mbuf, mbuf,
                                                            denom, N);
    edge_exp_kernel<<<blocks(TE), TPB, 0, stream>>>(ebuf, mbuf, denom, src, dst, E,
                                                    doubled);
    edge_alpha_kernel<<<blocks(TE), TPB, 0, stream>>>(ebuf, denom, src, dst, E,
                                                      doubled, alpha_dst);
    fill_zero_kernel<<<blocks((long long)N * 64), TPB, 0, stream>>>(obuf,
                                                                    (long long)N * 64);
    scatter_att_kernel<<<blocks((long long)TE * 64), TPB, 0, stream>>>(
        hbuf, alpha_dst, src, dst, E, doubled, obuf);
  };

  auto run_cgcn = [&](const float* feat, const float* Wf, const float* bf,
                      const float* Wg, const float* bg, const float* asrc,
                      const float* adst, const float* pref, float* rep,
                      float* alpha_dst) {
    // feats = l2norm(leaky_relu(feat @ Wf + bf)) into xbuf rows NU..N-1
    gemm128(feat, Wf, bf, xbuf + (size_t)NU * 64, NI, 0, 1);
    l2norm_kernel<<<(NI + 7) / 8, 256, 0, stream>>>(xbuf + (size_t)NU * 64,
                                                    xbuf + (size_t)NU * 64, NI);
    // pref = l2norm(pref) into xbuf rows 0..NU-1
    l2norm_kernel<<<(NU + 7) / 8, 256, 0, stream>>>(pref, xbuf, NU);
    for (int r = 0; r < 3; ++r) {
      gat_pass(Wg, bg, asrc, adst, 0, ebuf);
      pref_update_kernel<<<(NU + 7) / 8, 256, 0, stream>>>(xbuf, obuf, bg, NU);
    }
    gat_pass(Wg, bg, asrc, adst, 1, alpha_dst);
    rep_kernel<<<blocks((long long)N * 64), TPB, 0, stream>>>(xbuf, obuf, bg, rep,
                                                              (long long)N * 64);
  };

  run_cgcn(v_feat, Wf_v, bf_v, Wg_v, bg_v, asrc_v, adst_v, pref_v, repv, alphav);
  run_cgcn(a_feat, Wf_a, bf_a, Wg_a, bg_a, asrc_a, adst_a, pref_a, repa, alphaa);

  // edge fusion weights
  weight_kernel<<<blocks(2LL * E), TPB, 0, stream>>>(alphav, alphaa, msc, src, dst,
                                                     E, wedge);

  // EGCN: xn = l2norm(id_emb)
  l2norm_kernel<<<(N + 7) / 8, 256, 0, stream>>>(id_emb, xbuf, N);

  // h1 = leaky_relu(agg(xn) @ W1l + xn @ W1r + b1)
  fill_zero_kernel<<<blocks((long long)N * 64), TPB, 0, stream>>>(obuf,
                                                                  (long long)N * 64);
  scatter_sage_kernel<<<blocks(2LL * E * 64), TPB, 0, stream>>>(xbuf, wedge, src,
                                                                dst, E, obuf);
  gemm64(obuf, W1l, nullptr, h1buf, N, 0, 0);
  gemm64(xbuf, W1r, b1, h1buf, N, 1, 1);

  // h2 = leaky_relu(agg(h1) @ W2l + h1 @ W2r + b2)
  fill_zero_kernel<<<blocks((long long)N * 64), TPB, 0, stream>>>(obuf,
                                                                  (long long)N * 64);
  scatter_sage_kernel<<<blocks(2LL * E * 64), TPB, 0, stream>>>(h1buf, wedge, src,
                                                                dst, E, obuf);
  gemm64(obuf, W2l, nullptr, hbuf, N, 0, 0);
  gemm64(h1buf, W2r, b2, hbuf, N, 1, 1);

  // out = concat(xn + h1 + h2, rep_v, rep_a)
  final_out_kernel<<<blocks((long long)N * 64), TPB, 0, stream>>>(
      xbuf, h1buf, hbuf, repv, repa, (float*)d_out, (long long)N * 64);
}